// YOLOLayer_23398981829084
// MI455X (gfx1250) — compile-verified
//
#include <hip/hip_runtime.h>
#include <hip/hip_bf16.h>

typedef __attribute__((ext_vector_type(2))) float v2f;
typedef __attribute__((ext_vector_type(8))) float v8f;

#define N_CH        85
#define FS          80
#define PIX         6400      // 80*80
#define KIC         128
#define OCP         256       // output channels padded 255 -> 256
#define TILES_IMG   400       // 6400 / 16
#define NTILES      12800     // 32 * 400
#define STRIDE_F    8.0f

// GEMM mapping per wave (wave32):
//   A (16x4 f32, M=pixels, K=ic):  lane L holds M=L&15; V0 = K=k0+2*(L>>4), V1 = K+1
//   B (4x16  f32, K=ic, N=oc):     lane L holds N=L&15; V0 = K=k0+2*(L>>4), V1 = K+1
//   D (16x16 f32, M=pixels, N=oc): lane L, reg r -> M = r + 8*(L>>4), N = L&15
__global__ __launch_bounds__(256) void yolo_head_kernel(
    const float* __restrict__ xin,     // (32,128,80,80)
    const float* __restrict__ conv_w,  // (255,128)
    const float* __restrict__ conv_b,  // (255,)
    float* __restrict__ out,           // (32, 3*6400, 85)
    int total_waves)
{
    extern __shared__ float wlds[];    // Wt[ic][oc] : 128 x 256 f32 = 128 KB

    const int tid = threadIdx.x;

    // Cooperative weight transpose into LDS: thread t owns output channel t.
    {
        const int oc = tid;
        const bool valid = (oc < 255);
        #pragma unroll 4
        for (int ic = 0; ic < KIC; ++ic) {
            wlds[ic * OCP + oc] = valid ? conv_w[oc * KIC + ic] : 0.0f;
        }
    }
    __syncthreads();

    const int lane = tid & 31;
    const int ln   = lane & 15;   // N index (oc) / M index (pixel) for fragments
    const int hi   = lane >> 4;   // half-wave select (K offset of +2)
    const int wave = tid >> 5;
    const int gw   = blockIdx.x * 8 + wave;

    for (int t = gw; t < NTILES; t += total_waves) {
        const int b  = t / TILES_IMG;
        const int p0 = (t - b * TILES_IMG) * 16;          // pixel base within image
        const float* Xb = xin + (size_t)b * (KIC * PIX);

        // ---- Load full A panel: 16 pixels x 128 ic  (32 k-steps x float2 = 64 VGPRs)
        v2f A[32];
        {
            const float* xp = Xb + p0 + ln;
            #pragma unroll
            for (int ks = 0; ks < 32; ++ks) {
                const int k0 = 4 * ks + 2 * hi;
                A[ks].x = xp[(size_t)k0 * PIX];
                A[ks].y = xp[(size_t)(k0 + 1) * PIX];
            }
        }

        // ---- Loop over 16 oc-tiles of 16 channels, K=128 via 32 WMMAs each
        for (int ot = 0; ot < 16; ++ot) {
            v8f acc = {0.f, 0.f, 0.f, 0.f, 0.f, 0.f, 0.f, 0.f};
            const float* wp = wlds + (size_t)(2 * hi) * OCP + ot * 16 + ln;
            #pragma unroll
            for (int ks = 0; ks < 32; ++ks) {
                v2f bt;
                bt.x = wp[(size_t)(4 * ks)     * OCP];
                bt.y = wp[(size_t)(4 * ks + 1) * OCP];
                acc = __builtin_amdgcn_wmma_f32_16x16x4_f32(
                    /*neg_a=*/false, A[ks], /*neg_b=*/false, bt,
                    /*c_mod=*/(short)0, acc, /*reuse_a=*/false, /*reuse_b=*/false);
            }

            // ---- YOLO epilogue for this 16x16 tile
            const int   oc   = ot * 16 + ln;            // 0..255 (255 is padding)
            const int   a    = oc / 85;
            const int   c    = oc - a * 85;
            const float bias = conv_b[oc < 255 ? oc : 254];
            const bool  isw  = (c == 2);
            const bool  ish  = (c == 3);
            const bool  iswh = isw || ish;
            // anchors (grid units * stride 8 == original pixel anchors)
            const float anc  = isw ? (a == 0 ? 10.f : (a == 1 ? 16.f : 33.f))
                                   : (a == 0 ? 13.f : (a == 1 ? 30.f : 23.f));
            float* obase = out + ((size_t)(b * 3 + a) * PIX) * N_CH + c;

            #pragma unroll
            for (int r = 0; r < 8; ++r) {
                const int   p   = p0 + r + 8 * hi;      // pixel within image
                const float v   = acc[r] + bias;
                const float ex  = __expf(iswh ? v : -v);
                const float sig = 1.0f / (1.0f + ex);
                float res;
                if (c == 0) {
                    res = (sig + (float)(p % FS)) * STRIDE_F;   // x + grid-x, *stride
                } else if (c == 1) {
                    res = (sig + (float)(p / FS)) * STRIDE_F;   // y + grid-y, *stride
                } else if (iswh) {
                    res = ex * anc;                             // exp * anchor(px)
                } else {
                    res = sig;                                  // obj + classes
                }
                if (oc < 255) {
                    obase[(size_t)p * N_CH] = res;
                }
            }
        }
    }
}

extern "C" void kernel_launch(void* const* d_in, const int* in_sizes, int n_in,
                              void* d_out, int out_size, void* d_ws, size_t ws_size,
                              hipStream_t stream) {
    const float* xin    = (const float*)d_in[0];
    const float* conv_w = (const float*)d_in[1];
    const float* conv_b = (const float*)d_in[2];
    float*       out    = (float*)d_out;

    const int blocks = 400;                 // 3200 waves -> 4 pixel-tiles per wave
    const int total_waves = blocks * 8;
    const size_t shmem = (size_t)KIC * OCP * sizeof(float);   // 128 KB

    yolo_head_kernel<<<blocks, 256, shmem, stream>>>(xin, conv_w, conv_b, out, total_waves);
}